// DeTokenizer_14800457302188
// MI455X (gfx1250) — compile-verified
//
#include <hip/hip_runtime.h>
#include <cstdint>

// Problem constants (match reference)
constexpr int B = 8;
constexpr int L = 4096;
constexpr int D = 1024;
constexpr int M = 2048;
constexpr int NCHUNK = M / 16;

typedef float v2f __attribute__((ext_vector_type(2)));
typedef float v4f __attribute__((ext_vector_type(4)));
typedef float v8f __attribute__((ext_vector_type(8)));
typedef unsigned int u32x4 __attribute__((ext_vector_type(4)));
typedef int i32x4 __attribute__((ext_vector_type(4)));
typedef int i32x8 __attribute__((ext_vector_type(8)));

__device__ __forceinline__ float rl_f32(float v, int srclane) {
    return __int_as_float(__builtin_amdgcn_readlane(__float_as_int(v), srclane));
}

// Issue a TDM load of a 16x16 f32 tile (row stride D floats) from global
// address `gaddr` into LDS byte offset `lds_addr`. 2D tensor: groups 2/3 zero.
// 6-arg builtin on this toolchain:
//   (u32x4 g0, i32x8 g1, i32x4 g2, i32x4 g3, i32x8 extra, i32 cpol)
__device__ __forceinline__ void tdm_load_tile16x16(uint64_t gaddr, uint32_t lds_addr) {
    u32x4 g0;
    g0[0] = 1u;                                  // count=1, user descriptor
    g0[1] = lds_addr;                            // D#.lds_addr (bytes)
    g0[2] = (uint32_t)gaddr;                     // global_addr[31:0]
    g0[3] = (uint32_t)((gaddr >> 32) & 0x1FFFFFFu) | (2u << 30); // [56:32] | type=2
    i32x8 g1;
    g1[0] = (int)(2u << 16);                     // data_size=2 (4 bytes); wg_mask=0
    g1[1] = (int)(16u << 16);                    // tensor_dim0 = 16 (cols)
    g1[2] = (int)(16u << 16);                    // tensor_dim1 = 16 (rows)
    g1[3] = (int)(16u << 16);                    // tile_dim0  = 16
    g1[4] = (int)(16u);                          // tile_dim1  = 16, tile_dim2 = 0
    g1[5] = (int)D;                              // tensor_dim0_stride = 1024
    g1[6] = 0;
    g1[7] = 0;
    i32x4 z4 = {0, 0, 0, 0};
    i32x8 z8 = {0, 0, 0, 0, 0, 0, 0, 0};
    __builtin_amdgcn_tensor_load_to_lds(g0, g1, z4, z4, z8, 0);
}

// ---------------------------------------------------------------------------
// Kernel A: per-batch mask cumsum -> chunk_idx (B,L) and decay (B,M)
// One block (256 threads) per batch; each thread owns 16 consecutive tokens.
// ---------------------------------------------------------------------------
__global__ void detok_scan_kernel(const unsigned char* __restrict__ mask,
                                  const float* __restrict__ prob,
                                  float* __restrict__ decay,
                                  int* __restrict__ chunk_idx) {
    const int b = blockIdx.x;
    const int t = threadIdx.x;

    // init decay to 1.0 (positions with no token keep decay == 1)
    #pragma unroll
    for (int i = t; i < M; i += 256) decay[b * M + i] = 1.0f;

    // load 16 mask bytes (aligned uint4)
    const uint4* mrow = (const uint4*)(mask + (size_t)b * L);
    uint4 mv = mrow[t];
    unsigned int mw[4] = {mv.x, mv.y, mv.z, mv.w};

    int localCount = 0;
    #pragma unroll
    for (int e = 0; e < 16; ++e)
        localCount += (int)((mw[e >> 2] >> ((e & 3) * 8)) & 0xFFu ? 1 : 0);

    __shared__ int s[256];
    s[t] = localCount;
    __syncthreads();
    // Hillis-Steele inclusive scan over 256 threads
    #pragma unroll
    for (int off = 1; off < 256; off <<= 1) {
        int v = (t >= off) ? s[t - off] : 0;
        __syncthreads();
        s[t] += v;
        __syncthreads();
    }
    int run = s[t] - localCount; // exclusive prefix

    const float* prow = prob + (size_t)b * L;
    #pragma unroll
    for (int e = 0; e < 16; ++e) {
        int m = (int)((mw[e >> 2] >> ((e & 3) * 8)) & 0xFFu ? 1 : 0);
        run += m;
        int l = t * 16 + e;
        chunk_idx[(size_t)b * L + l] = run - 1;
        if (m) {
            float p = prow[l];
            float dv = 1.0f - p;
            dv = fminf(fmaxf(dv, 0.0f), 1.0f);
            decay[b * M + (run - 1)] = dv; // positions are unique: plain store
        }
    }
}

// ---------------------------------------------------------------------------
// Kernel B: chunked EMA scan via V_WMMA_F32_16X16X4_F32 with TDM staging.
// Block = 256 threads = 8 waves; wave w handles d-columns
// [blockIdx.y*128 + w*16, +16). Serial over 128 chunks of 16 timesteps.
//
// Per chunk: the 16x16 x-tile is DMA'd into a per-wave double-buffered LDS
// slice by the Tensor Data Mover, overlapped with the previous chunk's
// compute (s_wait_tensorcnt(1): in-order per-wave completion). Compute:
// lower-triangular decay-product matrix T (A-fragments held in named scalars
// so no alloca survives into scratch), b = (1-a)*x from LDS (B-fragments).
// The cross-chunk carry is folded into the K=0 element of the first
// B-fragment (P[t]*carry == T[t,0]*(a[0]*carry)), so the 4 chained K=4 f32
// WMMAs accumulate from zero. carry crosses chunks via ds_bpermute.
// ---------------------------------------------------------------------------
__global__ void __launch_bounds__(256, 1)
detok_ema_wmma_kernel(const float* __restrict__ hidden,
                      const float* __restrict__ decay,
                      const float* __restrict__ state,
                      const int* __restrict__ counts,
                      float* __restrict__ ema) {
    const int b    = blockIdx.x;
    const int wave = threadIdx.x >> 5;
    const int lane = threadIdx.x & 31;
    const int col  = lane & 15;  // d-column within tile / matrix N (and A row)
    const int hi   = lane >> 4;  // half-wave select
    const int d0   = blockIdx.y * 128 + wave * 16;

    const int cnt = counts[b];
    const float* hb = hidden + (size_t)b * M * D + d0;
    const float* db = decay  + (size_t)b * M;

    // store base: rows (c*16 + 8*hi + v), column col; advance 16 rows/chunk
    float* sb = ema + (size_t)b * M * D + (size_t)(8 * hi) * D + d0 + col;

    // per-wave double buffer: [buf][wave][t*16 + col], 16 KB total
    __shared__ float xbuf[2][8][256];
    const uint32_t lds0 = (uint32_t)(size_t)(&xbuf[0][wave][0]);
    const uint32_t lds1 = (uint32_t)(size_t)(&xbuf[1][wave][0]);

    // carry-in = initial state for this column (duplicated across both halves)
    float carry = state[(size_t)b * D + d0 + col];

    // prologue: DMA chunk 0 into buffer 0
    tdm_load_tile16x16((uint64_t)(size_t)hb, lds0);

    for (int c = 0; c < NCHUNK; ++c) {
        // ---- issue TDM for chunk c+1 into the other buffer, then wait for
        //      chunk c's tile (in-order per-wave tensor completion) ---------
        if (c + 1 < NCHUNK) {
            const float* nx = hb + (size_t)(c + 1) * 16 * D;
            tdm_load_tile16x16((uint64_t)(size_t)nx, ((c + 1) & 1) ? lds1 : lds0);
            __builtin_amdgcn_s_wait_tensorcnt((short)1);
        } else {
            __builtin_amdgcn_s_wait_tensorcnt((short)0);
        }

        // ---- broadcast the 16 decay values of this chunk to all lanes ----
        float aval = db[c * 16 + col]; // lanes 16..31 redundantly load same
        float a_loc[16];
        #pragma unroll
        for (int i = 0; i < 16; ++i) a_loc[i] = rl_f32(aval, i);

        // ---- per-lane row r=col of T: T[r,s] = prod_{u=s+1..r} a[u] ------
        float Trow[16];
        #pragma unroll
        for (int s2 = 0; s2 < 16; ++s2) Trow[s2] = (s2 == col) ? 1.0f : 0.0f;
        #pragma unroll
        for (int s2 = 14; s2 >= 0; --s2) {
            float nv = Trow[s2 + 1] * a_loc[s2 + 1];
            Trow[s2] = (s2 < col) ? nv : Trow[s2];
        }

        // ---- extract the 8 A-fragment values into NAMED SCALARS ----------
        // (an array of v2f here fails SROA and lands in scratch)
        // A 16x4 layout: vgpr0 = K=4j+2*hi, vgpr1 = K=4j+1+2*hi
        float af0x = hi ? Trow[2]  : Trow[0],  af0y = hi ? Trow[3]  : Trow[1];
        float af1x = hi ? Trow[6]  : Trow[4],  af1y = hi ? Trow[7]  : Trow[5];
        float af2x = hi ? Trow[10] : Trow[8],  af2y = hi ? Trow[11] : Trow[9];
        float af3x = hi ? Trow[14] : Trow[12], af3y = hi ? Trow[15] : Trow[13];

        // ---- 4 chained K=4 WMMAs from zero: acc = T @ b' -----------------
        // carry folds into b'[0]: b'[0] = (1-a[0])*x[0] + a[0]*carry (K=0
        // lives in the hi==0 half of the first Bf.x only).
        v8f acc = {};
        const float* lw = &xbuf[c & 1][wave][0];
        float cterm = hi ? 0.0f : a_loc[0] * carry;

        auto step = [&](float afx, float afy, float a0, float a1,
                        int t0, float extra) {
            float x0 = lw[t0 * 16 + col];        // bank-conflict-free
            float x1 = lw[(t0 + 1) * 16 + col];
            v2f Af, Bf;
            Af.x = afx; Af.y = afy;
            Bf.x = (1.0f - a0) * x0 + extra;
            Bf.y = (1.0f - a1) * x1;
            acc = __builtin_amdgcn_wmma_f32_16x16x4_f32(
                false, Af, false, Bf, (short)0, acc, false, false);
        };

        step(af0x, af0y, hi ? a_loc[2]  : a_loc[0],  hi ? a_loc[3]  : a_loc[1],
             0 + 2 * hi, cterm);
        step(af1x, af1y, hi ? a_loc[6]  : a_loc[4],  hi ? a_loc[7]  : a_loc[5],
             4 + 2 * hi, 0.0f);
        step(af2x, af2y, hi ? a_loc[10] : a_loc[8],  hi ? a_loc[11] : a_loc[9],
             8 + 2 * hi, 0.0f);
        step(af3x, af3y, hi ? a_loc[14] : a_loc[12], hi ? a_loc[15] : a_loc[13],
             12 + 2 * hi, 0.0f);

        // ---- store ema chunk; mask only in the boundary chunk ------------
        if (c * 16 + 16 <= cnt) {           // wave-uniform fast path
            #pragma unroll
            for (int v = 0; v < 8; ++v)
                sb[(size_t)v * D] = acc[v]; // immediate offsets v*4096
        } else {
            int rem = cnt - c * 16 - 8 * hi;
            #pragma unroll
            for (int v = 0; v < 8; ++v)
                sb[(size_t)v * D] = (v < rem) ? acc[v] : 0.0f;
        }
        sb += (size_t)16 * D;

        // ---- carry = h[15, col]; row 15 lives in acc[7] of lanes 16..31 --
        carry = __shfl(acc[7], 16 + col, 32);
    }
}

// ---------------------------------------------------------------------------
// Kernel C: output = residual + gather(ema, chunk_idx). Pure bandwidth,
// 128-bit (b128) loads/stores via ext-vector v4f. residual/out are streamed
// once -> non-temporal hints, so the 64 MB ema workspace stays resident in
// the 192 MB L2 for the gather.
// ---------------------------------------------------------------------------
__global__ void detok_out_kernel(const v4f* __restrict__ residual,
                                 const int* __restrict__ chunk_idx,
                                 const v4f* __restrict__ ema,
                                 v4f* __restrict__ out) {
    const int row = blockIdx.x;       // b*L + l
    const int b   = row >> 12;        // L = 4096
    const int t   = threadIdx.x;      // D/4 = 256
    const int idx = chunk_idx[row];

    v4f r = __builtin_nontemporal_load(&residual[(size_t)row * (D / 4) + t]);
    v4f e = {0.f, 0.f, 0.f, 0.f};
    if (idx >= 0)
        e = ema[((size_t)b * M + idx) * (D / 4) + t];
    v4f o = r + e;
    __builtin_nontemporal_store(o, &out[(size_t)row * (D / 4) + t]);
}

// ---------------------------------------------------------------------------
// Kernel D: new_state = counts>0 ? ema[counts-1] : state
// ---------------------------------------------------------------------------
__global__ void detok_state_kernel(const float* __restrict__ ema,
                                   const float* __restrict__ state,
                                   const int* __restrict__ counts,
                                   float* __restrict__ out_state) {
    const int i = blockIdx.x * blockDim.x + threadIdx.x; // B*D
    const int b = i >> 10;  // D = 1024
    const int d = i & 1023;
    const int c = counts[b];
    out_state[i] = (c > 0) ? ema[((size_t)b * M + (c - 1)) * D + d] : state[i];
}

// ---------------------------------------------------------------------------
extern "C" void kernel_launch(void* const* d_in, const int* in_sizes, int n_in,
                              void* d_out, int out_size, void* d_ws, size_t ws_size,
                              hipStream_t stream) {
    const float*         hidden   = (const float*)d_in[0];         // (B,M,D)
    const float*         residual = (const float*)d_in[1];         // (B,L,D)
    const unsigned char* mask     = (const unsigned char*)d_in[2]; // (B,L) bool
    const float*         prob     = (const float*)d_in[3];         // (B,L)
    const int*           counts   = (const int*)d_in[4];           // (B,)
    const float*         state    = (const float*)d_in[5];         // (B,D)

    float* out       = (float*)d_out;                 // (B,L,D)
    float* out_state = out + (size_t)B * L * D;       // (B,D)

    char*  ws        = (char*)d_ws;
    float* decay     = (float*)ws;                    // B*M f32   (64 KB)
    int*   chunk_idx = (int*)(ws + (64 << 10));       // B*L i32   (128 KB)
    float* ema       = (float*)(ws + (256 << 10));    // B*M*D f32 (64 MB)

    detok_scan_kernel<<<B, 256, 0, stream>>>(mask, prob, decay, chunk_idx);
    detok_ema_wmma_kernel<<<dim3(B, D / 128), 256, 0, stream>>>(
        hidden, decay, state, counts, ema);
    detok_out_kernel<<<B * L, 256, 0, stream>>>(
        (const v4f*)residual, chunk_idx, (const v4f*)ema, (v4f*)out);
    detok_state_kernel<<<(B * D) / 256, 256, 0, stream>>>(
        ema, state, counts, out_state);
}